// AdditiveAttention_34617436405767
// MI455X (gfx1250) — compile-verified
//
#include <hip/hip_runtime.h>
#include <hip/hip_bf16.h>
#include <math.h>

#define B_ 32
#define S_ 2048
#define H_ 1024
#define SSPLIT 16
#define SCHUNK (S_ / SSPLIT)   // 128

typedef __attribute__((ext_vector_type(2))) float v2f;
typedef __attribute__((ext_vector_type(4))) float v4f;
typedef __attribute__((ext_vector_type(8))) float v8f;

// ---------------------------------------------------------------------------
// Projection GEMM via V_WMMA_F32_16X16X4_F32. One wave per 16x16 output tile.
// BMODE 0: B-matrix is W indexed [k][n] (row-major KxN)       -> out = A @ W
// BMODE 1: B-matrix element [k][n] taken from W[n][k] (W^T)   -> out = A @ W^T
// ISA f32 A 16x4 layout: lanes 0-15 hold (K=0,K=1) in (v0,v1); lanes 16-31 hold (K=2,K=3).
// ISA f32 B 4x16 layout mirrors: lanes 0-15 = N col for K rows (0,1); lanes 16-31 for (2,3).
// C/D: lane l holds column N=(l&15); VGPR j holds row M = j + 8*(l>>4).
// ---------------------------------------------------------------------------
template <int BMODE, bool ADD_BIAS>
__global__ __launch_bounds__(32) void proj_wmma_kernel(
    const float* __restrict__ A, const float* __restrict__ W,
    const float* __restrict__ bias, float* __restrict__ out) {
  const int lane = threadIdx.x & 31;
  const int half = lane >> 4;   // 0: lanes 0-15, 1: lanes 16-31
  const int l16  = lane & 15;
  const int m_base = blockIdx.y * 16;
  const int n_base = blockIdx.x * 16;
  const int mrow = m_base + l16;
  const int ncol = n_base + l16;

  v8f acc = {};
  for (int k = 0; k < H_; k += 4) {
    const int ka = k + 2 * half;
    v2f a = *(const v2f*)(A + (size_t)mrow * H_ + ka);
    v2f b;
    if (BMODE == 0) {
      b.x = W[(size_t)ka * H_ + ncol];
      b.y = W[(size_t)(ka + 1) * H_ + ncol];
    } else {
      b = *(const v2f*)(W + (size_t)ncol * H_ + ka);
    }
    acc = __builtin_amdgcn_wmma_f32_16x16x4_f32(
        /*neg_a=*/false, a, /*neg_b=*/false, b,
        /*c_mod=*/(short)0, acc, /*reuse_a=*/false, /*reuse_b=*/false);
  }

  const float bv = ADD_BIAS ? bias[ncol] : 0.0f;
#pragma unroll
  for (int j = 0; j < 8; ++j) {
    const int orow = m_base + j + 8 * half;
    out[(size_t)orow * H_ + ncol] = acc[j] + bv;
  }
}

// ---------------------------------------------------------------------------
// scores[b,s] = key[b,s,:] . qk[b,:]  (+ mask). One wave per (b,s) row.
// Per iteration the wave loads 32 contiguous float4 = 512B of key (coalesced).
// key is single-use: non-temporal loads to avoid polluting L2.
// ---------------------------------------------------------------------------
__global__ __launch_bounds__(256) void scores_kernel(
    const float* __restrict__ key, const float* __restrict__ qk,
    const unsigned char* __restrict__ mask, float* __restrict__ scores) {
  const int wave = threadIdx.x >> 5;
  const int lane = threadIdx.x & 31;
  const int row  = blockIdx.x * 8 + wave;   // in [0, B*S)
  const int b    = row >> 11;               // row / S_
  const v4f* kp = (const v4f*)(key + (size_t)row * H_);
  const v4f* qp = (const v4f*)(qk + (size_t)b * H_);
  float acc = 0.0f;
#pragma unroll
  for (int i = 0; i < 8; ++i) {
    v4f kv = __builtin_nontemporal_load(kp + lane + i * 32);
    v4f qv = qp[lane + i * 32];
    acc = fmaf(kv.x, qv.x, acc);
    acc = fmaf(kv.y, qv.y, acc);
    acc = fmaf(kv.z, qv.z, acc);
    acc = fmaf(kv.w, qv.w, acc);
  }
#pragma unroll
  for (int off = 16; off; off >>= 1) acc += __shfl_xor(acc, off, 32);
  if (lane == 0) {
    // reference: scores - 1e9*(1 - mask); the constant q.bk term cancels in softmax
    scores[row] = acc + (mask[row] ? 0.0f : -1e9f);
  }
}

// ---------------------------------------------------------------------------
// Row softmax over S=2048. One 256-thread block per batch row, 8 elems/thread.
// ---------------------------------------------------------------------------
__global__ __launch_bounds__(256) void softmax_kernel(
    const float* __restrict__ scores, float* __restrict__ probs) {
  __shared__ float red[256];
  const int b = blockIdx.x;
  const int t = threadIdx.x;
  const float* src = scores + (size_t)b * S_;
  float v[8];
  float lm = -INFINITY;
#pragma unroll
  for (int i = 0; i < 8; ++i) { v[i] = src[t + i * 256]; lm = fmaxf(lm, v[i]); }
  red[t] = lm; __syncthreads();
  for (int o = 128; o; o >>= 1) {
    if (t < o) red[t] = fmaxf(red[t], red[t + o]);
    __syncthreads();
  }
  const float mx = red[0]; __syncthreads();
  float ls = 0.0f;
#pragma unroll
  for (int i = 0; i < 8; ++i) { v[i] = __expf(v[i] - mx); ls += v[i]; }
  red[t] = ls; __syncthreads();
  for (int o = 128; o; o >>= 1) {
    if (t < o) red[t] += red[t + o];
    __syncthreads();
  }
  const float inv = 1.0f / red[0];
#pragma unroll
  for (int i = 0; i < 8; ++i) probs[(size_t)b * S_ + t + i * 256] = v[i] * inv;
}

// ---------------------------------------------------------------------------
// context partial: each block owns (b, s-chunk, 256 h-columns); one thread per
// h column accumulates over SCHUNK s. value is single-use -> NT loads.
// Deterministic two-phase (no float atomics).
// ---------------------------------------------------------------------------
__global__ __launch_bounds__(256) void context_partial_kernel(
    const float* __restrict__ probs, const float* __restrict__ value,
    float* __restrict__ partial) {
  __shared__ float pl[SCHUNK];
  const int t = threadIdx.x;
  const int h = blockIdx.x * 256 + t;
  const int chunk = blockIdx.y;
  const int b = blockIdx.z;
  const int s0 = chunk * SCHUNK;
  if (t < SCHUNK) pl[t] = probs[(size_t)b * S_ + s0 + t];
  __syncthreads();
  const float* vp = value + ((size_t)b * S_ + s0) * H_ + h;
  float acc = 0.0f;
#pragma unroll 8
  for (int s = 0; s < SCHUNK; ++s) {
    acc = fmaf(pl[s], __builtin_nontemporal_load(vp + (size_t)s * H_), acc);
  }
  partial[((size_t)b * SSPLIT + chunk) * H_ + h] = acc;
}

__global__ __launch_bounds__(256) void context_reduce_kernel(
    const float* __restrict__ partial, float* __restrict__ out) {
  const int idx = blockIdx.x * 256 + threadIdx.x;  // b*H + h
  const int b = idx >> 10;        // / H_
  const int h = idx & (H_ - 1);
  float acc = 0.0f;
#pragma unroll
  for (int c = 0; c < SSPLIT; ++c)
    acc += partial[((size_t)b * SSPLIT + c) * H_ + h];
  out[idx] = acc;
}

// ---------------------------------------------------------------------------
extern "C" void kernel_launch(void* const* d_in, const int* in_sizes, int n_in,
                              void* d_out, int out_size, void* d_ws, size_t ws_size,
                              hipStream_t stream) {
  const float* query = (const float*)d_in[0];
  const float* key   = (const float*)d_in[1];
  const float* value = (const float*)d_in[2];
  const unsigned char* mask = (const unsigned char*)d_in[3];  // bool array, 1B/elem
  const float* Wq = (const float*)d_in[4];
  const float* bq = (const float*)d_in[5];
  const float* Wk = (const float*)d_in[6];
  // d_in[7] = bk: q.bk is constant per row -> cancels in softmax, never needed.

  float* ws    = (float*)d_ws;
  float* q_ws  = ws;                         // B*H
  float* qk_ws = q_ws + B_ * H_;             // B*H
  float* sc_ws = qk_ws + B_ * H_;            // B*S
  float* pr_ws = sc_ws + B_ * S_;            // B*S
  float* pt_ws = pr_ws + B_ * S_;            // B*SSPLIT*H
  float* out   = (float*)d_out;

  // 1) q = query @ Wq + bq        (WMMA f32, 2x64 tiles)
  proj_wmma_kernel<0, true><<<dim3(H_ / 16, B_ / 16), 32, 0, stream>>>(query, Wq, bq, q_ws);
  // 2) qk = q @ Wk^T              (WMMA f32)
  proj_wmma_kernel<1, false><<<dim3(H_ / 16, B_ / 16), 32, 0, stream>>>(q_ws, Wk, nullptr, qk_ws);
  // 3) scores = key . qk (+mask)  (streams 256MB of key)
  scores_kernel<<<(B_ * S_) / 8, 256, 0, stream>>>(key, qk_ws, mask, sc_ws);
  // 4) softmax rows
  softmax_kernel<<<B_, 256, 0, stream>>>(sc_ws, pr_ws);
  // 5) context = probs @ value    (streams 256MB of value), deterministic 2-phase
  context_partial_kernel<<<dim3(H_ / 256, SSPLIT, B_), 256, 0, stream>>>(pr_ws, value, pt_ws);
  context_reduce_kernel<<<(B_ * H_) / 256, 256, 0, stream>>>(pt_ws, out);
}